// BiLSTM_CRF_31937376813689
// MI455X (gfx1250) — compile-verified
//
#include <hip/hip_runtime.h>
#include <math.h>

// ---------------------------------------------------------------------------
// Problem constants (fp32 BiLSTM-CRF, batch=1)
// ---------------------------------------------------------------------------
#define TT    2048      // sequence length
#define EE    768       // embedding dim
#define HH    512       // hidden per direction
#define GG    2048      // 4*HH gate width
#define KK    7         // number of tags
#define STARTT 5
#define STOPP  6
#define NEGV  (-10000.0f)

// LSTM recurrence partitioning
#define NWG      16                 // workgroups per direction
#define ROWS     128                // gate rows per WG (4 gates x 32 elems)
#define WPAD     516                // LDS row stride (512 + 4) breaks bank conflicts
#define LSTM_SMEM_FLOATS (ROWS * WPAD + HH + ROWS + 32)
#define LSTM_SMEM_BYTES  (LSTM_SMEM_FLOATS * sizeof(float))

typedef float v2f __attribute__((ext_vector_type(2)));
typedef float v8f __attribute__((ext_vector_type(8)));

// ---------------------------------------------------------------------------
// Kernel 1: pre[dir][t][j] = embeds[t][:] . W_ih[j][:] + (b_ih[j]+b_hh[j])
// fp32 WMMA 16x16x4. Block = 128 threads (4 waves). Each wave owns a 16x64
// output strip (4 x v8f accumulators, A fragment reused 4x per k-step).
// Grid: (TT/16, GG/256, 2 dirs).
// ---------------------------------------------------------------------------
__global__ __launch_bounds__(128) void pre_gemm_kernel(
    const float* __restrict__ X,                         // [TT][EE]
    const float* __restrict__ Wf, const float* __restrict__ Wb,   // [GG][EE]
    const float* __restrict__ bihf, const float* __restrict__ bhhf,
    const float* __restrict__ bihb, const float* __restrict__ bhhb,
    float* __restrict__ pre)                             // [2][TT][GG]
{
    const int dir  = blockIdx.z;
    const float* __restrict__ W   = dir ? Wb   : Wf;
    const float* __restrict__ bih = dir ? bihb : bihf;
    const float* __restrict__ bhh = dir ? bhhb : bhhf;
    float* __restrict__ out = pre + (size_t)dir * TT * GG;

    const int lane  = threadIdx.x & 31;
    const int wave  = threadIdx.x >> 5;
    const int mbase = blockIdx.x * 16;
    const int nbase = blockIdx.y * 256 + wave * 64;
    const int l15   = lane & 15;
    const int hi    = lane >> 4;          // 0 for lanes 0-15, 1 for 16-31
    const int koff  = hi << 1;            // K sub-offset {0,2} per ISA A/B layout

    // A fragment source: row m = mbase+l15, contiguous K pair at kb+koff
    const float* arow = X + (size_t)(mbase + l15) * EE + koff;
    // B fragment source: B[k][n] = W[n][k]; same contiguous-K-pair addressing
    const float* brow[4];
#pragma unroll
    for (int tn = 0; tn < 4; ++tn)
        brow[tn] = W + (size_t)(nbase + tn * 16 + l15) * EE + koff;

    v8f acc[4];
#pragma unroll
    for (int tn = 0; tn < 4; ++tn) acc[tn] = (v8f){};

    for (int kb = 0; kb < EE; kb += 4) {
        v2f a = *reinterpret_cast<const v2f*>(arow + kb);
#pragma unroll
        for (int tn = 0; tn < 4; ++tn) {
            v2f b = *reinterpret_cast<const v2f*>(brow[tn] + kb);
            acc[tn] = __builtin_amdgcn_wmma_f32_16x16x4_f32(
                false, a, false, b, (short)0, acc[tn], false, false);
        }
    }

    // Epilogue: C/D layout -> VGPR r holds rows (r, r+8), N = lane&15
    const int rofs = mbase + 8 * hi;
#pragma unroll
    for (int tn = 0; tn < 4; ++tn) {
        const int n = nbase + tn * 16 + l15;
        const float bias = bih[n] + bhh[n];
#pragma unroll
        for (int r = 0; r < 8; ++r)
            out[(size_t)(rofs + r) * GG + n] = acc[tn][r] + bias;
    }
}

// ---------------------------------------------------------------------------
// Kernel 2: sequential LSTM recurrence, distributed across 16 WGPs per
// direction (grid = (16, 2), 128 threads/WG).
//
// WG w owns hidden elements k in [w*32, w*32+32) and, crucially, the four
// gate rows {k, 512+k, 1024+k, 1536+k} for those k — so each WG updates its
// h/c slice with NO cross-WG z exchange. Its 128 W_hh rows (256 KB, padded
// stride 516 to avoid 64-bank conflicts) are staged in LDS ONCE; every step
// reads weights only from LDS. Per step the WGs share the 512-float h vector
// through L2 (the hs[t] row itself) and synchronize with a monotonic
// device-scope atomic barrier (release fence -> atomicAdd -> acquire spin).
// Gate order i,f,g,o (PyTorch split of z into quarters).
// ---------------------------------------------------------------------------
__device__ __forceinline__ float sigm(float x) { return 1.0f / (1.0f + expf(-x)); }

__global__ __launch_bounds__(128) void lstm_kernel(
    const float* __restrict__ pre_all,                    // [2][TT][GG]
    const float* __restrict__ Whhf, const float* __restrict__ Whhb,  // [GG][HH]
    const float* __restrict__ h0, const float* __restrict__ c0,      // [2][HH]
    float* __restrict__ hs_all,                           // [2][TT][HH]
    unsigned int* __restrict__ ctr)                       // [2] step counters
{
    extern __shared__ float smem[];
    float* Wl   = smem;                       // [ROWS][WPAD]
    float* h_sh = smem + ROWS * WPAD;         // [HH], 16B aligned
    float* z_sh = h_sh + HH;                  // [ROWS]
    float* c_sh = z_sh + ROWS;                // [32]

    const int w     = blockIdx.x;             // 0..NWG-1
    const int dir   = blockIdx.y;             // 0..1
    const int tid   = threadIdx.x;            // 0..127
    const int kbase = w * 32;
    const int g     = tid >> 5;               // gate 0..3
    const int e     = tid & 31;               // element within slice
    const int j     = g * HH + kbase + e;     // global gate row owned by tid

    const float* __restrict__ pre = pre_all + (size_t)dir * TT * GG;
    const float* __restrict__ Whh = dir ? Whhb : Whhf;
    float* __restrict__ hs = hs_all + (size_t)dir * TT * HH;
    unsigned int* myctr = ctr + dir;

    // Stage this WG's 128 W_hh rows into LDS (row d == tid, one-time cost)
    {
        const float4* src = reinterpret_cast<const float4*>(Whh + (size_t)j * HH);
        float4* dst = reinterpret_cast<float4*>(Wl + (size_t)tid * WPAD);
        for (int c = 0; c < HH / 4; ++c) dst[c] = src[c];
    }
    if (tid < 32) c_sh[tid] = c0[dir * HH + kbase + tid];
    __syncthreads();

    float4* h4v = reinterpret_cast<float4*>(h_sh);
    const float4* wrow = reinterpret_cast<const float4*>(Wl + (size_t)tid * WPAD);

    for (int s = 0; s < TT; ++s) {
        const int t = dir ? (TT - 1 - s) : s;

        // h source: h0 for step 0, else the hs row all WGs published last step
        const float* hrow = (s == 0) ? (h0 + dir * HH)
                                     : (hs + (size_t)(dir ? t + 1 : t - 1) * HH);
        const float  pv   = pre[(size_t)t * GG + j];     // issue early (L2)
        const float4 hv   = reinterpret_cast<const float4*>(hrow)[tid];
        h4v[tid] = hv;
        __syncthreads();

        // one 512-long dot per thread, weights from LDS, h broadcast from LDS
        float acc = 0.0f;
#pragma unroll 8
        for (int kk = 0; kk < HH / 4; ++kk) {
            const float4 wv = wrow[kk];
            const float4 hh = h4v[kk];
            acc = fmaf(wv.x, hh.x, fmaf(wv.y, hh.y, fmaf(wv.z, hh.z, fmaf(wv.w, hh.w, acc))));
        }
        z_sh[tid] = pv + acc;
        __syncthreads();

        if (tid < 32) {
            const float zi = z_sh[tid];
            const float zf = z_sh[32 + tid];
            const float zg = z_sh[64 + tid];
            const float zo = z_sh[96 + tid];
            const float c  = sigm(zf) * c_sh[tid] + sigm(zi) * tanhf(zg);
            const float h  = sigm(zo) * tanhf(c);
            c_sh[tid] = c;
            hs[(size_t)t * HH + kbase + tid] = h;        // publish our h slice
        }
        __threadfence();                                  // push stores to device scope
        __syncthreads();

        if (tid == 0) {
            __hip_atomic_fetch_add(myctr, 1u, __ATOMIC_RELEASE, __HIP_MEMORY_SCOPE_AGENT);
            const unsigned int target = (unsigned int)(NWG * (s + 1));
            while (__hip_atomic_load(myctr, __ATOMIC_ACQUIRE, __HIP_MEMORY_SCOPE_AGENT) < target)
                __builtin_amdgcn_s_sleep(1);
        }
        __syncthreads();
    }
}

// ---------------------------------------------------------------------------
// Kernel 3: feats[t][k] = concat(hs_f[t], hs_b[t]) . W_out[k] + b_out[k]
// ---------------------------------------------------------------------------
__global__ __launch_bounds__(256) void feats_kernel(
    const float* __restrict__ hs_all,    // [2][TT][HH]
    const float* __restrict__ Wout,      // [KK][1024]
    const float* __restrict__ bout,      // [KK]
    float* __restrict__ feats)           // [TT][KK]
{
    const int idx = blockIdx.x * blockDim.x + threadIdx.x;
    if (idx >= TT * KK) return;
    const int t = idx / KK, k = idx % KK;

    const float4* hf = reinterpret_cast<const float4*>(hs_all + (size_t)t * HH);
    const float4* hb = reinterpret_cast<const float4*>(hs_all + (size_t)TT * HH + (size_t)t * HH);
    const float4* wf = reinterpret_cast<const float4*>(Wout + (size_t)k * 1024);
    const float4* wb = wf + HH / 4;

    float s = bout[k];
#pragma unroll 4
    for (int i = 0; i < HH / 4; ++i) {
        const float4 a = hf[i], wv = wf[i];
        s = fmaf(a.x, wv.x, fmaf(a.y, wv.y, fmaf(a.z, wv.z, fmaf(a.w, wv.w, s))));
    }
#pragma unroll 4
    for (int i = 0; i < HH / 4; ++i) {
        const float4 a = hb[i], wv = wb[i];
        s = fmaf(a.x, wv.x, fmaf(a.y, wv.y, fmaf(a.z, wv.z, fmaf(a.w, wv.w, s))));
    }
    feats[idx] = s;
}

// ---------------------------------------------------------------------------
// Kernel 4: Viterbi. One wave. Transitions, feats chunks and ALL
// backpointers live in LDS (~22 KB), so the 2048-step backtrack chain
// never leaves the WGP. First-index argmax matches jnp.argmax.
// out[0] = path_score; out[1..TT] = path tags (as floats).
// ---------------------------------------------------------------------------
#define VCHUNK 256

__global__ __launch_bounds__(32) void viterbi_kernel(
    const float* __restrict__ feats,     // [TT][KK]
    const float* __restrict__ trans,     // [KK][KK]
    float* __restrict__ out)             // [1 + TT]
{
    __shared__ float tr[KK * KK];
    __shared__ float fv[KK];
    __shared__ float fbuf[VCHUNK * KK];
    __shared__ unsigned char bp[TT * KK];

    const int lane = threadIdx.x;
    for (int i = lane; i < KK * KK; i += 32) tr[i] = trans[i];
    if (lane < KK) fv[lane] = (lane == STARTT) ? 0.0f : NEGV;
    __syncthreads();

    for (int tc = 0; tc < TT; tc += VCHUNK) {
        for (int i = lane; i < VCHUNK * KK; i += 32) fbuf[i] = feats[tc * KK + i];
        __syncthreads();
        for (int tt = 0; tt < VCHUNK; ++tt) {
            float newv = 0.0f;
            if (lane < KK) {
                float best = fv[0] + tr[lane * KK + 0];
                int   arg  = 0;
                for (int kp = 1; kp < KK; ++kp) {
                    const float v = fv[kp] + tr[lane * KK + kp];
                    if (v > best) { best = v; arg = kp; }   // first-max wins ties
                }
                newv = best + fbuf[tt * KK + lane];
                bp[(tc + tt) * KK + lane] = (unsigned char)arg;
            }
            __syncthreads();
            if (lane < KK) fv[lane] = newv;
            __syncthreads();
        }
    }

    if (lane == 0) {
        float best = fv[0] + tr[STOPP * KK + 0];
        int   bt   = 0;
        for (int kp = 1; kp < KK; ++kp) {
            const float v = fv[kp] + tr[STOPP * KK + kp];
            if (v > best) { best = v; bt = kp; }
        }
        out[0] = best;
        int tag = bt;
        for (int t = TT - 1; t >= 0; --t) {
            out[1 + t] = (float)tag;            // path[T-1]=best; path[t]=bp[t+1][path[t+1]]
            tag = bp[t * KK + tag];
        }
    }
}

// ---------------------------------------------------------------------------
// Host launcher. Workspace layout (floats):
//   pre   : 2*TT*GG   = 8,388,608
//   hs    : 2*TT*HH   = 2,097,152
//   feats : TT*KK     =    14,336
//   ctr   : 2 x u32   (barrier counters, reset every launch)
// total ~42 MB.
// ---------------------------------------------------------------------------
extern "C" void kernel_launch(void* const* d_in, const int* in_sizes, int n_in,
                              void* d_out, int out_size, void* d_ws, size_t ws_size,
                              hipStream_t stream) {
    (void)in_sizes; (void)n_in; (void)out_size; (void)ws_size;

    const float* embeds = (const float*)d_in[0];
    const float* h0     = (const float*)d_in[1];
    const float* c0     = (const float*)d_in[2];
    const float* Wih_f  = (const float*)d_in[3];
    const float* Whh_f  = (const float*)d_in[4];
    const float* bih_f  = (const float*)d_in[5];
    const float* bhh_f  = (const float*)d_in[6];
    const float* Wih_b  = (const float*)d_in[7];
    const float* Whh_b  = (const float*)d_in[8];
    const float* bih_b  = (const float*)d_in[9];
    const float* bhh_b  = (const float*)d_in[10];
    const float* Wout   = (const float*)d_in[11];
    const float* bout   = (const float*)d_in[12];
    const float* trans  = (const float*)d_in[13];

    float* out   = (float*)d_out;
    float* ws    = (float*)d_ws;
    float* pre   = ws;
    float* hs    = ws + (size_t)2 * TT * GG;
    float* feats = hs + (size_t)2 * TT * HH;
    unsigned int* ctr = (unsigned int*)(feats + (size_t)TT * KK);

    // Allow >64KB dynamic LDS for the recurrence kernel (CDNA5: 320KB/WG).
    // Unconditional + idempotent: no static state, deterministic every call,
    // and not a stream operation so it is graph-capture safe.
    (void)hipFuncSetAttribute((const void*)lstm_kernel,
                              hipFuncAttributeMaxDynamicSharedMemorySize,
                              (int)LSTM_SMEM_BYTES);

    // Reset barrier counters (deterministic across graph replays)
    (void)hipMemsetAsync(ctr, 0, 2 * sizeof(unsigned int), stream);

    // 1) Input GEMMs for both directions (WMMA f32 16x16x4)
    dim3 g1(TT / 16, GG / 256, 2);
    pre_gemm_kernel<<<g1, 128, 0, stream>>>(embeds, Wih_f, Wih_b,
                                            bih_f, bhh_f, bih_b, bhh_b, pre);

    // 2) Recurrence: 16 WGPs per direction, LDS-resident weights,
    //    device-scope atomic barrier per step
    lstm_kernel<<<dim3(NWG, 2), 128, LSTM_SMEM_BYTES, stream>>>(
        pre, Whh_f, Whh_b, h0, c0, hs, ctr);

    // 3) Emission features
    feats_kernel<<<(TT * KK + 255) / 256, 256, 0, stream>>>(hs, Wout, bout, feats);

    // 4) Viterbi decode
    viterbi_kernel<<<1, 32, 0, stream>>>(feats, trans, out);
}